// Capsule_51307679318384
// MI455X (gfx1250) — compile-verified
//
#include <hip/hip_runtime.h>

typedef __attribute__((ext_vector_type(2))) float v2f;
typedef __attribute__((ext_vector_type(8))) float v8f;

#define B_    32
#define N_    2048
#define DIN   16
#define K_    32
#define DOUT  16
#define WAVES 4
#define JB    256
#define NBLK  (N_ / JB)

// Compute u[:, k, j, :] as two 16x16 f32 tiles (b=0..15, b=16..31) via
// four chained V_WMMA_F32_16X16X4_F32 per tile (K = D_in = 16 = 4 steps of 4).
// A layout (16x4 f32): lane L -> row m=L%16; vgpr0/1 = K = 2*half, 2*half+1.
// B layout (4x16 f32): lane L -> col n=L%16; vgpr0/1 = K = 2*half, 2*half+1.
__device__ __forceinline__ void wmma_u_tiles(const float* __restrict__ x,
                                             const float* __restrict__ wkj,
                                             int j, int m, int half,
                                             v8f& acc0, v8f& acc1) {
#pragma unroll
  for (int s = 0; s < 4; ++s) {
    v2f bt;
    bt.x = wkj[(4 * s + 2 * half) * DOUT + m];
    bt.y = wkj[(4 * s + 2 * half + 1) * DOUT + m];
    const float* xa = x + (((size_t)m * N_) + j) * DIN + 4 * s + 2 * half;
    const float* xb = xa + (size_t)16 * N_ * DIN;   // batch tile 1 (b = m+16)
    v2f a0 = *(const v2f*)xa;
    v2f a1 = *(const v2f*)xb;
    acc0 = __builtin_amdgcn_wmma_f32_16x16x4_f32(false, a0, false, bt,
                                                 (short)0, acc0, false, false);
    acc1 = __builtin_amdgcn_wmma_f32_16x16x4_f32(false, a1, false, bt,
                                                 (short)0, acc1, false, false);
  }
}

__global__ void init_accum_kernel(float* __restrict__ ws) {
  int i = blockIdx.x * blockDim.x + threadIdx.x;
  if (i < 2 * B_ * K_ * DOUT) ws[i] = 0.0f;   // zero t (16384) and s (16384)
}

// Pass 1: t[b,k,:] = sum_j x[b,j,:] @ w[k,j,:,:]  -- accumulate entirely in the
// WMMA C operand across the j loop, then LDS cross-wave reduce + f32 atomics.
__global__ __launch_bounds__(WAVES * 32)
void pass_t(const float* __restrict__ x, const float* __restrict__ w,
            float* __restrict__ t_out) {
  __shared__ float red[WAVES][32][16];
  int k    = blockIdx.x / NBLK;
  int j0   = (blockIdx.x % NBLK) * JB;
  int wave = threadIdx.x >> 5;
  int lane = threadIdx.x & 31;
  int m = lane & 15, half = lane >> 4;

  v8f acc0 = {}, acc1 = {};
  for (int j = j0 + wave; j < j0 + JB; j += WAVES) {
    const float* wkj = w + (((size_t)k * N_) + j) * (DIN * DOUT);
    __builtin_prefetch(wkj + (size_t)WAVES * DIN * DOUT + lane * 8, 0, 3);
    wmma_u_tiles(x, wkj, j, m, half, acc0, acc1);
  }
#pragma unroll
  for (int v = 0; v < 8; ++v) {
    red[wave][lane][v]     = acc0[v];
    red[wave][lane][8 + v] = acc1[v];
  }
  __syncthreads();
  if (wave == 0) {
#pragma unroll
    for (int idx = 0; idx < 16; ++idx) {
      float sum = red[0][lane][idx] + red[1][lane][idx] +
                  red[2][lane][idx] + red[3][lane][idx];
      int v = idx & 7, tile = idx >> 3;
      int b = tile * 16 + half * 8 + v;              // C layout: M = v + 8*half
      atomicAdd(&t_out[((size_t)b * K_ + k) * DOUT + m], sum);
    }
  }
}

// Pass 2: logits c_raw[b,k,j] = <u_{bkj}, t_{bk}> / sqrt(16)
__global__ __launch_bounds__(WAVES * 32)
void pass_logits(const float* __restrict__ x, const float* __restrict__ w,
                 const float* __restrict__ t_in, float* __restrict__ c_raw) {
  int k    = blockIdx.x / NBLK;
  int j0   = (blockIdx.x % NBLK) * JB;
  int wave = threadIdx.x >> 5;
  int lane = threadIdx.x & 31;
  int m = lane & 15, half = lane >> 4;

  float tv0[8], tv1[8];
#pragma unroll
  for (int v = 0; v < 8; ++v) {
    tv0[v] = t_in[((size_t)(half * 8 + v) * K_ + k) * DOUT + m];
    tv1[v] = t_in[((size_t)(16 + half * 8 + v) * K_ + k) * DOUT + m];
  }
  for (int j = j0 + wave; j < j0 + JB; j += WAVES) {
    const float* wkj = w + (((size_t)k * N_) + j) * (DIN * DOUT);
    __builtin_prefetch(wkj + (size_t)WAVES * DIN * DOUT + lane * 8, 0, 3);
    v8f acc0 = {}, acc1 = {};
    wmma_u_tiles(x, wkj, j, m, half, acc0, acc1);
#pragma unroll
    for (int v = 0; v < 8; ++v) {
      float p0 = acc0[v] * tv0[v];   // dot over z: reduce across lanes of half
      p0 += __shfl_xor(p0, 1, 32); p0 += __shfl_xor(p0, 2, 32);
      p0 += __shfl_xor(p0, 4, 32); p0 += __shfl_xor(p0, 8, 32);
      float p1 = acc1[v] * tv1[v];
      p1 += __shfl_xor(p1, 1, 32); p1 += __shfl_xor(p1, 2, 32);
      p1 += __shfl_xor(p1, 4, 32); p1 += __shfl_xor(p1, 8, 32);
      if (m == v) {
        int b0 = half * 8 + v;
        c_raw[((size_t)b0 * K_ + k) * N_ + j]        = p0 * 0.25f;
        c_raw[((size_t)(b0 + 16) * K_ + k) * N_ + j] = p1 * 0.25f;
      }
    }
  }
}

// Pass 3: softmax over k (in-place: each thread owns c_raw[b, 0..K-1, j]) + bias
__global__ void pass_softmax(float* __restrict__ c, const float* __restrict__ bias) {
  int idx = blockIdx.x * blockDim.x + threadIdx.x;   // over B*N
  if (idx >= B_ * N_) return;
  int b = idx / N_, j = idx % N_;
  float lc[K_];
  float mx = -1e30f;
#pragma unroll
  for (int k = 0; k < K_; ++k) {
    lc[k] = c[((size_t)b * K_ + k) * N_ + j];
    mx = fmaxf(mx, lc[k]);
  }
  float sum = 0.0f;
#pragma unroll
  for (int k = 0; k < K_; ++k) { lc[k] = __expf(lc[k] - mx); sum += lc[k]; }
  float inv = 1.0f / sum;
#pragma unroll
  for (int k = 0; k < K_; ++k)
    c[((size_t)b * K_ + k) * N_ + j] = lc[k] * inv + bias[(size_t)k * N_ + j];
}

// Pass 4: s[b,k,:] = sum_j c[b,k,j] * u[b,k,j,:]  (recompute u, broadcast c via shfl)
__global__ __launch_bounds__(WAVES * 32)
void pass_s(const float* __restrict__ x, const float* __restrict__ w,
            const float* __restrict__ c, float* __restrict__ s_out) {
  __shared__ float red[WAVES][32][16];
  int k    = blockIdx.x / NBLK;
  int j0   = (blockIdx.x % NBLK) * JB;
  int wave = threadIdx.x >> 5;
  int lane = threadIdx.x & 31;
  int m = lane & 15, half = lane >> 4;

  v8f sa0 = {}, sa1 = {};
  for (int j = j0 + wave; j < j0 + JB; j += WAVES) {
    const float* wkj = w + (((size_t)k * N_) + j) * (DIN * DOUT);
    __builtin_prefetch(wkj + (size_t)WAVES * DIN * DOUT + lane * 8, 0, 3);
    v8f acc0 = {}, acc1 = {};
    wmma_u_tiles(x, wkj, j, m, half, acc0, acc1);
    float cs = c[((size_t)lane * K_ + k) * N_ + j];  // lane L holds c[b=L,k,j]
#pragma unroll
    for (int v = 0; v < 8; ++v) {
      float c0 = __shfl(cs, half * 8 + v, 32);       // b = v + 8*half
      float c1 = __shfl(cs, 16 + half * 8 + v, 32);  // b = 16 + v + 8*half
      sa0[v] += acc0[v] * c0;
      sa1[v] += acc1[v] * c1;
    }
  }
#pragma unroll
  for (int v = 0; v < 8; ++v) {
    red[wave][lane][v]     = sa0[v];
    red[wave][lane][8 + v] = sa1[v];
  }
  __syncthreads();
  if (wave == 0) {
#pragma unroll
    for (int idx = 0; idx < 16; ++idx) {
      float sum = red[0][lane][idx] + red[1][lane][idx] +
                  red[2][lane][idx] + red[3][lane][idx];
      int v = idx & 7, tile = idx >> 3;
      int b = tile * 16 + half * 8 + v;
      atomicAdd(&s_out[((size_t)b * K_ + k) * DOUT + m], sum);
    }
  }
}

// Pass 5: efficient squash
__global__ void pass_squash(const float* __restrict__ s_in, float* __restrict__ out) {
  int idx = blockIdx.x * blockDim.x + threadIdx.x;   // over B*K
  if (idx >= B_ * K_) return;
  const float* sp = s_in + (size_t)idx * DOUT;
  float sv[DOUT];
  float nsq = 0.0f;
#pragma unroll
  for (int z = 0; z < DOUT; ++z) { sv[z] = sp[z]; nsq += sv[z] * sv[z]; }
  float n = sqrtf(nsq);
  float scale = (1.0f - 1.0f / (__expf(n) + 1e-20f)) / (n + 1e-20f);
  float* op = out + (size_t)idx * DOUT;
#pragma unroll
  for (int z = 0; z < DOUT; ++z) op[z] = sv[z] * scale;
}

extern "C" void kernel_launch(void* const* d_in, const int* in_sizes, int n_in,
                              void* d_out, int out_size, void* d_ws, size_t ws_size,
                              hipStream_t stream) {
  const float* x    = (const float*)d_in[0];   // [B, N, DIN]
  const float* w    = (const float*)d_in[1];   // [K, N, DIN, DOUT]
  const float* bias = (const float*)d_in[2];   // [K, N, 1]
  float* out = (float*)d_out;                  // [B, K, DOUT]

  // Workspace: t (16384 f), s (16384 f), c (B*K*N = 2097152 f)  -> ~8.5 MB
  float* ws    = (float*)d_ws;
  float* t_buf = ws;
  float* s_buf = ws + (size_t)B_ * K_ * DOUT;
  float* c_buf = ws + 2 * (size_t)B_ * K_ * DOUT;

  init_accum_kernel<<<(2 * B_ * K_ * DOUT + 255) / 256, 256, 0, stream>>>(ws);
  pass_t      <<<K_ * NBLK, WAVES * 32, 0, stream>>>(x, w, t_buf);
  pass_logits <<<K_ * NBLK, WAVES * 32, 0, stream>>>(x, w, t_buf, c_buf);
  pass_softmax<<<(B_ * N_ + 255) / 256, 256, 0, stream>>>(c_buf, bias);
  pass_s      <<<K_ * NBLK, WAVES * 32, 0, stream>>>(x, w, c_buf, s_buf);
  pass_squash <<<(B_ * K_ + 255) / 256, 256, 0, stream>>>(s_buf, out);
}